// AutoSVS_317827580767
// MI455X (gfx1250) — compile-verified
//
#include <hip/hip_runtime.h>

#define Bsz 16
#define Jn  96
#define Nn  384
#define Tt  4096
#define Dd  256
#define TR  32    // t-rows per workgroup (two 16-row WMMA M-tiles)
#define NP  386   // half-row stride; 2*NP = 772 == 4 (mod 64) -> conflict-free A reads
#define RS  (2 * NP)   // full row stride: [w1 row | w2 row]

#define SMEM_BYTES ((TR * RS + 3 * TR) * sizeof(float))

typedef float v2f __attribute__((ext_vector_type(2)));
typedef float v8f __attribute__((ext_vector_type(8)));

struct NoteParams {
    float cnt, m1, s1, q1, m2, s2, q2;
    int   wstart, wend, pad;
};

// ---------------------------------------------------------------------------
// Setup: per-batch scans (J cumsum, phoneme-run -> note map) + per-(b,n)
// gaussian parameter precompute. Tiny vs. the main kernel.
// ---------------------------------------------------------------------------
__global__ void AutoSVS_setup_kernel(const float* __restrict__ duration,
                                     const int* __restrict__ phoneme_seq,
                                     const int* __restrict__ note_duration,
                                     const unsigned char* __restrict__ is_multi,
                                     NoteParams* __restrict__ params)
{
    int b = blockIdx.x;
    __shared__ int s_start[Jn], s_dur[Jn], s_ws[Jn], s_we[Jn];
    __shared__ int s_nidx[Nn];
    __shared__ unsigned char s_valid[Nn];

    if (threadIdx.x == 0) {
        int csum = 0;
        for (int j = 0; j < Jn; ++j) {
            int nd = note_duration[b * Jn + j];
            s_dur[j] = nd; s_start[j] = csum; csum += nd;
        }
        for (int j = 0; j < Jn; ++j) {
            int prev = (j > 0)      ? s_dur[j - 1] : 0; if (prev > 10) prev = 10;
            int nxt  = (j < Jn - 1) ? s_dur[j + 1] : 0; if (nxt  > 10) nxt  = 10;
            s_ws[j] = s_start[j] - prev;
            s_we[j] = s_start[j] + s_dur[j] + nxt;
        }
        int g = -1, prevv = 0;
        for (int n = 0; n < Nn; ++n) {
            int v = (phoneme_seq[b * Nn + n] > 1) ? 1 : 0;
            if (v && !prevv) ++g;
            prevv = v;
            int gi = g < 0 ? 0 : (g > Jn - 1 ? Jn - 1 : g);
            s_nidx[n]  = gi;
            s_valid[n] = (unsigned char)v;
        }
    }
    __syncthreads();

    for (int n = threadIdx.x; n < Nn; n += blockDim.x) {
        int j = s_nidx[n];
        const float* du = duration + ((size_t)b * Nn + n) * 6;
        float phd  = (float)s_dur[j];
        float m1p  = fminf(fmaxf(du[0], -0.25f), 1.25f);
        float std1 = __expf(du[1]) + 1.0f;
        float amp1 = __expf(du[2]);
        float m2p  = fminf(fmaxf(du[3], -0.25f), 1.25f);
        float std2 = __expf(du[4]) + 1.0f;
        float amp2 = __expf(du[5]);
        bool v    = s_valid[n] != 0;
        bool mult = is_multi[b * Nn + n] != 0;

        NoteParams p;
        p.cnt = (float)s_start[j];
        p.m1 = m1p * phd;
        p.s1 = v ? (amp1 / std1) : 0.0f;
        p.q1 = 1.0f / (2.0f * std1 * std1);
        p.m2 = m2p * phd;
        p.s2 = (v && mult) ? (amp2 / std2) : 0.0f;
        p.q2 = 1.0f / (2.0f * std2 * std2);
        p.wstart = s_ws[j];
        p.wend   = s_we[j];
        p.pad    = 0;
        params[b * Nn + n] = p;
    }
}

// ---------------------------------------------------------------------------
// Fused main kernel: per (b, 32-row t-tile)
//   phase 1: w1,w2 [32 x 384] -> interleaved LDS rows [w1 | w2],
//            a = w1+w2+alignment cached in VGPRs, L1 row sums via ds_add_f32
//   phase 2: align_n = a / max(norm,1e-12) (non-temporal, from registers)
//   phase 3: out = (w1 @ x1 + w2 @ x2) / max(norm,1e-10) with
//            v_wmma_f32_16x16x4_f32; each B-frag feeds TWO M-tiles; all four
//            A-frags per k-step use immediate ds offsets off one base.
// ---------------------------------------------------------------------------
__global__ __launch_bounds__(256)
void AutoSVS_main_kernel(const float* __restrict__ x,          // [B,N,2D]
                         const float* __restrict__ alignment,  // [B,T,N]
                         const NoteParams* __restrict__ params,
                         float* __restrict__ out,              // [B,T,D]
                         float* __restrict__ align_out)        // [B,T,N]
{
    extern __shared__ float smem[];
    float* s_w      = smem;                 // TR rows of [w1(NP) | w2(NP)]
    float* s_rowsum = smem + TR * RS;       // TR
    float* s_inv10  = s_rowsum + TR;        // TR
    float* s_inv12  = s_inv10 + TR;         // TR

    const int b    = blockIdx.y;
    const int t0   = blockIdx.x * TR;
    const int tid  = threadIdx.x;
    const int lane = tid & 31;
    const int wv   = tid >> 5;              // wave id 0..7

    if (tid < TR) s_rowsum[tid] = 0.0f;
    __syncthreads();

    // ----- phase 1: gaussians -> LDS; a = w1+w2+alignment cached in VGPRs -----
    // wave wv owns rows {wv, wv+8, wv+16, wv+24}; lane owns n = lane + 32k.
    float areg[4][12];
    float psum[4] = {0.f, 0.f, 0.f, 0.f};
    #pragma unroll
    for (int k = 0; k < 12; ++k) {
        const int n = lane + 32 * k;
        NoteParams p = params[b * Nn + n];
        #pragma unroll
        for (int i = 0; i < 4; ++i) {
            const int r = wv + 8 * i;
            const int t = t0 + r;
            float fin = (float)t - p.cnt;
            bool inw  = (t >= p.wstart) && (t < p.wend);
            float d1 = fin - p.m1;
            float e1 = fminf(-(d1 * d1) * p.q1, 20.0f);
            float w1 = inw ? p.s1 * __expf(e1) : 0.0f;
            float d2 = fin - p.m2;
            float e2 = fminf(-(d2 * d2) * p.q2, 20.0f);
            float w2 = inw ? p.s2 * __expf(e2) : 0.0f;
            s_w[r * RS + n]      = w1;
            s_w[r * RS + NP + n] = w2;
            float al = __builtin_nontemporal_load(
                alignment + ((size_t)b * Tt + t) * Nn + n);
            float a = w1 + w2 + al;
            areg[i][k] = a;
            psum[i] += fabsf(a);
        }
    }
    #pragma unroll
    for (int i = 0; i < 4; ++i) atomicAdd(&s_rowsum[wv + 8 * i], psum[i]);
    __syncthreads();

    if (tid < TR) {
        float s = s_rowsum[tid];
        s_inv10[tid] = 1.0f / fmaxf(s, 1e-10f);
        s_inv12[tid] = 1.0f / fmaxf(s, 1e-12f);
    }
    __syncthreads();

    // ----- phase 2: normalized alignment straight from registers -----
    #pragma unroll
    for (int i = 0; i < 4; ++i) {
        const int r = wv + 8 * i;
        float inv = s_inv12[r];
        float* aout = align_out + ((size_t)b * Tt + t0 + r) * Nn + lane;
        #pragma unroll
        for (int k = 0; k < 12; ++k)
            __builtin_nontemporal_store(areg[i][k] * inv, aout + 32 * k);
    }

    // ----- phase 3: fp32 WMMA GEMM; each B-frag reused by two M-tiles -----
    const int hi = lane >> 4;               // half-wave: K offset selector
    const int lo = lane & 15;               // A: row M / B,C: col
    const float* xb = x + (size_t)b * Nn * (2 * Dd);
    const float* abase = s_w + lo * RS;     // this lane's A-row base

    for (int sub = 0; sub < 2; ++sub) {
        const int d0 = (wv * 2 + sub) * 16;
        const float* xc1 = xb + d0 + lo;        // x[..., :D] column
        const float* xc2 = xb + Dd + d0 + lo;   // x[..., D:] column
        v8f acc0 = {0.f, 0.f, 0.f, 0.f, 0.f, 0.f, 0.f, 0.f};   // rows t0+0..15
        v8f acc1 = {0.f, 0.f, 0.f, 0.f, 0.f, 0.f, 0.f, 0.f};   // rows t0+16..31

        for (int n = 0; n < Nn; n += 4) {
            const int kk = n + 2 * hi;          // per-half-wave K base (even)
            v2f b1, b2;
            // B 4x16 fp32 layout: lane holds col N=lo, VGPR v -> K = kk+v
            b1.x = xc1[(size_t)kk * (2 * Dd)];  b1.y = xc1[(size_t)(kk + 1) * (2 * Dd)];
            b2.x = xc2[(size_t)kk * (2 * Dd)];  b2.y = xc2[(size_t)(kk + 1) * (2 * Dd)];
            // A 16x4 fp32 layout: lane holds row M=lo(+16), VGPR v -> K = kk+v
            // four frags at immediate offsets {0, NP, 16*RS, 16*RS+NP} from one base
            v2f a1  = *(const v2f*)(abase + kk);
            v2f a2  = *(const v2f*)(abase + NP + kk);
            v2f a1b = *(const v2f*)(abase + 16 * RS + kk);
            v2f a2b = *(const v2f*)(abase + 16 * RS + NP + kk);
            acc0 = __builtin_amdgcn_wmma_f32_16x16x4_f32(false, a1,  false, b1,
                                                         (short)0, acc0, false, false);
            acc0 = __builtin_amdgcn_wmma_f32_16x16x4_f32(false, a2,  false, b2,
                                                         (short)0, acc0, false, false);
            acc1 = __builtin_amdgcn_wmma_f32_16x16x4_f32(false, a1b, false, b1,
                                                         (short)0, acc1, false, false);
            acc1 = __builtin_amdgcn_wmma_f32_16x16x4_f32(false, a2b, false, b2,
                                                         (short)0, acc1, false, false);
        }

        // C/D layout: VGPR i -> row i (lanes 0-15) / row i+8 (lanes 16-31), col = lo
        float* orow = out + ((size_t)b * Tt + t0) * Dd + d0 + lo;
        #pragma unroll
        for (int i = 0; i < 8; ++i) {
            int row = i + hi * 8;
            __builtin_nontemporal_store(acc0[i] * s_inv10[row],
                                        orow + (size_t)row * Dd);
            __builtin_nontemporal_store(acc1[i] * s_inv10[row + 16],
                                        orow + (size_t)(row + 16) * Dd);
        }
    }
}

extern "C" void kernel_launch(void* const* d_in, const int* in_sizes, int n_in,
                              void* d_out, int out_size, void* d_ws, size_t ws_size,
                              hipStream_t stream)
{
    const float*         x             = (const float*)d_in[0];
    const float*         duration      = (const float*)d_in[1];
    const float*         alignment     = (const float*)d_in[2];
    const int*           phoneme_seq   = (const int*)d_in[3];
    const int*           note_duration = (const int*)d_in[4];
    const unsigned char* is_multi      = (const unsigned char*)d_in[5];

    float* out       = (float*)d_out;                       // [B,T,D]
    float* align_out = out + (size_t)Bsz * Tt * Dd;         // [B,T,N]
    NoteParams* params = (NoteParams*)d_ws;                 // B*N*40 bytes

    (void)hipFuncSetAttribute((const void*)AutoSVS_main_kernel,
                              hipFuncAttributeMaxDynamicSharedMemorySize,
                              (int)SMEM_BYTES);

    AutoSVS_setup_kernel<<<Bsz, 128, 0, stream>>>(duration, phoneme_seq,
                                                  note_duration, is_multi, params);
    AutoSVS_main_kernel<<<dim3(Tt / TR, Bsz), 256, SMEM_BYTES, stream>>>(
        x, alignment, params, out, align_out);
}